// LegendreActivation_8409545966396
// MI455X (gfx1250) — compile-verified
//
#include <hip/hip_runtime.h>

typedef __attribute__((ext_vector_type(2))) float v2f;
typedef __attribute__((ext_vector_type(4))) float v4f;
typedef __attribute__((ext_vector_type(8))) float v8f;

#define IN_DIM  1024
#define OUT_DIM 1024
#define NPOLY   9      // Legendre degrees 0..8

// tanh via hardware transcendentals: tanh(y) = 1 - 2/(exp2(2y*log2e)+1)
// Saturates correctly at +-1 for large |y| (exp2 -> inf or 0).
__device__ __forceinline__ float fast_tanh(float y) {
    float e = __builtin_amdgcn_exp2f(y * 2.8853900817779268f); // 2*log2(e)
    return 1.0f - 2.0f * __builtin_amdgcn_rcpf(e + 1.0f);
}

__global__ __launch_bounds__(256)
void legendre_fused_kernel(const float* __restrict__ x,
                           const float* __restrict__ coeffs,   // [OUT_DIM][9] row-major
                           const float* __restrict__ scale_p,  // 1 element
                           float* __restrict__ out)            // [B][OUT_DIM]
{
    __shared__ float S[16 * 16];   // S[local_row][k], k 0..8 valid, 9..15 zero pad

    const int tid  = threadIdx.x;
    const int lane = tid & 31;
    const int wave = tid >> 5;     // 8 waves
    const int b0   = blockIdx.x * 16;

    // zero the LDS tile (256 threads, 256 elements)
    S[tid] = 0.0f;
    __syncthreads();

    float s = scale_p[0];
    s = fminf(fmaxf(s, 0.1f), 2.0f);

    // ---------- Phase 1: per-row Legendre sums S[b, 0..8] ----------
    // wave handles rows (2*wave) and (2*wave+1) of this 16-row tile
    for (int r = 0; r < 2; ++r) {
        const int row = 2 * wave + r;
        const float* xrow = x + (size_t)(b0 + row) * IN_DIM;

        float acc[NPOLY];
#pragma unroll
        for (int k = 0; k < NPOLY; ++k) acc[k] = 0.0f;

        // 1024 elems / 32 lanes = 32 per lane, as 8 coalesced float4 loads
#pragma unroll
        for (int j = 0; j < 8; ++j) {
            v4f xv = *(const v4f*)(xrow + j * 128 + lane * 4);
#pragma unroll
            for (int e = 0; e < 4; ++e) {
                float t  = fast_tanh(xv[e] * s);
                float p0 = 1.0f;
                float p1 = t;
                acc[0] += p0;
                acc[1] += p1;
#pragma unroll
                for (int n = 2; n <= 8; ++n) {
                    // Bonnet: Pn = ((2n-1)/n) x P_{n-1} - ((n-1)/n) P_{n-2}
                    const float c1 = (2.0f * n - 1.0f) / (float)n;
                    const float c2 = (n - 1.0f) / (float)n;
                    float pn = c1 * t * p1 - c2 * p0;
                    acc[n] += pn;          // |t|<=1 => |Pn|<=1, clip is a no-op
                    p0 = p1;
                    p1 = pn;
                }
            }
        }

        // wave32 butterfly reduction over the input dimension
#pragma unroll
        for (int k = 0; k < NPOLY; ++k) {
#pragma unroll
            for (int off = 16; off > 0; off >>= 1)
                acc[k] += __shfl_xor(acc[k], off, 32);
        }
        if (lane == 0) {
#pragma unroll
            for (int k = 0; k < NPOLY; ++k) S[row * 16 + k] = acc[k];
        }
    }
    __syncthreads();

    // ---------- Phase 2: out tile = S(16x12) x coeffsT(12x16cols) via WMMA ----------
    const int mn    = lane & 15;          // M for A, N for B/D
    const int khalf = (lane >> 4) * 2;    // lanes<16 -> k{0,1}; lanes>=16 -> k{2,3}
    const int mrow  = (lane >> 4) * 8;    // D: M = vgpr + (lane>=16 ? 8 : 0)

    // A fragments (shared by all 8 column tiles this wave owns): 3 K-chunks of 4
    v2f afrag[3];
#pragma unroll
    for (int c = 0; c < 3; ++c) {
        afrag[c][0] = S[mn * 16 + 4 * c + khalf + 0];
        afrag[c][1] = S[mn * 16 + 4 * c + khalf + 1];
    }

    // 64 column tiles of 16; each wave does 8
#pragma unroll
    for (int j = 0; j < 8; ++j) {
        const int ct = wave * 8 + j;
        const int o  = ct * 16 + mn;      // output column for this lane

        v2f bfrag[3];
#pragma unroll
        for (int c = 0; c < 3; ++c) {
            const int k0 = 4 * c + khalf;
            bfrag[c][0] = (k0     < NPOLY) ? coeffs[o * NPOLY + k0    ] : 0.0f;
            bfrag[c][1] = (k0 + 1 < NPOLY) ? coeffs[o * NPOLY + k0 + 1] : 0.0f;
        }

        v8f d = {};
#pragma unroll
        for (int c = 0; c < 3; ++c) {
            // (neg_a, A, neg_b, B, c_mod, C, reuse_a, reuse_b)
            d = __builtin_amdgcn_wmma_f32_16x16x4_f32(
                    false, afrag[c], false, bfrag[c], (short)0, d, false, false);
        }

#pragma unroll
        for (int v = 0; v < 8; ++v)
            out[(size_t)(b0 + mrow + v) * OUT_DIM + ct * 16 + mn] = d[v];
    }
}

extern "C" void kernel_launch(void* const* d_in, const int* in_sizes, int n_in,
                              void* d_out, int out_size, void* d_ws, size_t ws_size,
                              hipStream_t stream) {
    (void)in_sizes; (void)n_in; (void)out_size; (void)d_ws; (void)ws_size;
    const float* x      = (const float*)d_in[0];
    const float* coeffs = (const float*)d_in[1];
    const float* scale  = (const float*)d_in[2];
    float* out          = (float*)d_out;

    const int row_tiles = 4096 / 16;   // 256 blocks, 256 threads (8 wave32s) each
    legendre_fused_kernel<<<row_tiles, 256, 0, stream>>>(x, coeffs, scale, out);
}